// sageconv_7086696039141
// MI455X (gfx1250) — compile-verified
//
#include <hip/hip_runtime.h>

typedef __attribute__((ext_vector_type(2))) float v2f;
typedef __attribute__((ext_vector_type(8))) float v8f;

#define DFEAT 128
#define LDSS  130   // padded LDS row stride (floats): 8B-aligned, bank-conflict-friendly

// ---------------- zero workspace (acc + deg) ----------------
__global__ void zero_ws_kernel(float4* ws4, int n4) {
    int i = blockIdx.x * blockDim.x + threadIdx.x;
    if (i < n4) ws4[i] = make_float4(0.f, 0.f, 0.f, 0.f);
}

// ---------------- edge scatter: one wave32 per edge ----------------
// lane l handles features [4l, 4l+4): float4 gather from feat[src], 4 f32
// global atomics into acc[dst]. lane 0 bumps deg[dst].
__global__ __launch_bounds__(256)
void scatter_edges_kernel(const float4* __restrict__ feat4,
                          const int* __restrict__ src,
                          const int* __restrict__ dst,
                          float* __restrict__ acc,
                          float* __restrict__ deg, int E) {
    int wave = threadIdx.x >> 5;
    int lane = threadIdx.x & 31;
    int e = blockIdx.x * 8 + wave;
    if (e >= E) return;
    int s = src[e];
    int d = dst[e];
    float4 v = feat4[(long)s * (DFEAT / 4) + lane];
    float* p = acc + (long)d * DFEAT + lane * 4;
    atomicAdd(p + 0, v.x);
    atomicAdd(p + 1, v.y);
    atomicAdd(p + 2, v.z);
    atomicAdd(p + 3, v.w);
    if (lane == 0) atomicAdd(deg + d, 1.0f);
}

// ---------------- fused dual GEMM + bias epilogue ----------------
// Block = 256 threads (8 waves). Block owns 64 node rows; wave w owns output
// columns [16w, 16w+16). Each wave keeps 4 accumulator tiles (4 x 16 rows).
// out = feat @ Wself^T + (acc/deg) @ Wneigh^T + bias, all f32 WMMA.
__global__ __launch_bounds__(256)
void gemm_out_kernel(const float* __restrict__ feat,
                     const float* __restrict__ acc,
                     const float* __restrict__ deg,
                     const float* __restrict__ Wself,
                     const float* __restrict__ Wneigh,
                     const float* __restrict__ bias,
                     float* __restrict__ out, int N) {
    __shared__ float sF[64 * LDSS];
    __shared__ float sN[64 * LDSS];

    const int tid  = threadIdx.x;
    const int row0 = blockIdx.x * 64;

    // Stage 64x128 tiles of feat and h_neigh (=acc/max(deg,1)) into LDS.
    for (int i = tid; i < 64 * (DFEAT / 4); i += 256) {
        int r  = i >> 5;              // 0..63
        int c4 = i & 31;              // float4 column
        int g  = row0 + r;
        if (g >= N) g = N - 1;        // clamp (tail rows masked at store)
        float4 f = ((const float4*)feat)[(long)g * (DFEAT / 4) + c4];
        float4 a = ((const float4*)acc)[(long)g * (DFEAT / 4) + c4];
        float rd = 1.0f / fmaxf(deg[g], 1.0f);
        int b = r * LDSS + c4 * 4;
        sF[b + 0] = f.x; sF[b + 1] = f.y; sF[b + 2] = f.z; sF[b + 3] = f.w;
        sN[b + 0] = a.x * rd; sN[b + 1] = a.y * rd;
        sN[b + 2] = a.z * rd; sN[b + 3] = a.w * rd;
    }
    __syncthreads();

    const int wave   = tid >> 5;
    const int lane   = tid & 31;
    const int lane16 = lane & 15;
    const int khi    = (lane < 16) ? 0 : 2;   // ISA 16x4 f32 A/B lane->K mapping
    const int ncol   = wave * 16 + lane16;    // this lane's output column

    float bv = bias[ncol];
    v8f c0 = {bv, bv, bv, bv, bv, bv, bv, bv};
    v8f c1 = c0, c2 = c0, c3 = c0;

    const float* Ws = Wself  + (long)ncol * DFEAT;
    const float* Wn = Wneigh + (long)ncol * DFEAT;

    for (int kk = 0; kk < DFEAT / 4; ++kk) {
        const int k = kk * 4 + khi;
        // B fragments: B[k..k+1, ncol] = W[ncol, k..k+1] (hot in WGP$/L2)
        v2f bs = *(const v2f*)(Ws + k);
        v2f bn = *(const v2f*)(Wn + k);
        v2f a;
        // t = 0..3 row tiles, B fragments reused 4x each
        a = *(const v2f*)(&sF[(0 * 16 + lane16) * LDSS + k]);
        c0 = __builtin_amdgcn_wmma_f32_16x16x4_f32(false, a, false, bs, (short)0, c0, false, false);
        a = *(const v2f*)(&sN[(0 * 16 + lane16) * LDSS + k]);
        c0 = __builtin_amdgcn_wmma_f32_16x16x4_f32(false, a, false, bn, (short)0, c0, false, false);

        a = *(const v2f*)(&sF[(1 * 16 + lane16) * LDSS + k]);
        c1 = __builtin_amdgcn_wmma_f32_16x16x4_f32(false, a, false, bs, (short)0, c1, false, false);
        a = *(const v2f*)(&sN[(1 * 16 + lane16) * LDSS + k]);
        c1 = __builtin_amdgcn_wmma_f32_16x16x4_f32(false, a, false, bn, (short)0, c1, false, false);

        a = *(const v2f*)(&sF[(2 * 16 + lane16) * LDSS + k]);
        c2 = __builtin_amdgcn_wmma_f32_16x16x4_f32(false, a, false, bs, (short)0, c2, false, false);
        a = *(const v2f*)(&sN[(2 * 16 + lane16) * LDSS + k]);
        c2 = __builtin_amdgcn_wmma_f32_16x16x4_f32(false, a, false, bn, (short)0, c2, false, false);

        a = *(const v2f*)(&sF[(3 * 16 + lane16) * LDSS + k]);
        c3 = __builtin_amdgcn_wmma_f32_16x16x4_f32(false, a, false, bs, (short)0, c3, false, false);
        a = *(const v2f*)(&sN[(3 * 16 + lane16) * LDSS + k]);
        c3 = __builtin_amdgcn_wmma_f32_16x16x4_f32(false, a, false, bn, (short)0, c3, false, false);
    }

    // Store: C/D layout -> VGPR v holds row (v + (lane<16?0:8)) of the 16-row tile.
    const int rsel = (lane < 16) ? 0 : 8;
#pragma unroll
    for (int v = 0; v < 8; ++v) {
        int r0 = row0 + 0 * 16 + v + rsel;
        int r1 = row0 + 1 * 16 + v + rsel;
        int r2 = row0 + 2 * 16 + v + rsel;
        int r3 = row0 + 3 * 16 + v + rsel;
        if (r0 < N) out[(long)r0 * DFEAT + ncol] = c0[v];
        if (r1 < N) out[(long)r1 * DFEAT + ncol] = c1[v];
        if (r2 < N) out[(long)r2 * DFEAT + ncol] = c2[v];
        if (r3 < N) out[(long)r3 * DFEAT + ncol] = c3[v];
    }
}

extern "C" void kernel_launch(void* const* d_in, const int* in_sizes, int n_in,
                              void* d_out, int out_size, void* d_ws, size_t ws_size,
                              hipStream_t stream) {
    const float* feat   = (const float*)d_in[0];
    const float* Wself  = (const float*)d_in[1];
    const float* Wneigh = (const float*)d_in[2];
    const float* bias   = (const float*)d_in[3];
    const int*   src    = (const int*)d_in[4];
    const int*   dst    = (const int*)d_in[5];
    float*       out    = (float*)d_out;

    const int N = in_sizes[0] / DFEAT;
    const int E = in_sizes[4];

    float* acc = (float*)d_ws;                 // N * 128 f32
    float* deg = acc + (long)N * DFEAT;        // N f32

    // 1) zero accumulator + degree
    int n4 = (N * (DFEAT + 1) + 3) / 4;
    zero_ws_kernel<<<(n4 + 255) / 256, 256, 0, stream>>>((float4*)d_ws, n4);

    // 2) edge gather + atomic scatter-add (one wave per edge)
    scatter_edges_kernel<<<(E + 7) / 8, 256, 0, stream>>>(
        (const float4*)feat, src, dst, acc, deg, E);

    // 3) fused mean-normalize + dual WMMA GEMM + bias
    gemm_out_kernel<<<(N + 63) / 64, 256, 0, stream>>>(
        feat, acc, deg, Wself, Wneigh, bias, out, N);
}